// CapLayer2_22728966930493
// MI455X (gfx1250) — compile-verified
//
#include <hip/hip_runtime.h>
#include <hip/hip_bf16.h>

typedef __attribute__((ext_vector_type(2))) float v2f;
typedef __attribute__((ext_vector_type(8))) float v8f;

#define BSZ   64
#define CIN   256
#define NI    1024          // I = 32*32 spatial positions
#define NJ    10            // output capsules
#define ND    32            // capsule dim
#define NO    (NJ * ND)     // 320 conv output channels
#define OTILES (NO / 16)    // 20
#define IGRPS  (NI / 64)    // 16 groups of 64 columns (4 x 16-wide WMMA tiles)
#define WAVES_PER_BLOCK 8

// ---------------------------------------------------------------------------
// Kernel 1: pred[b, o, i] = sum_c W[o,c] * x[b,c,i] + bias[o]
// One wave32 computes a 16(M=o) x 64(N=i) tile: 4 independent
// V_WMMA_F32_16X16X4_F32 accumulation chains sharing one A operand,
// accumulating K=256 in steps of 4.
//
// f32 WMMA layouts (wave32):
//   A (16x4, MxK):  lane&15 = M;  VGPR0 holds K = (lane>>4)*2, VGPR1 = K+1
//   B (4x16, KxN):  lane&15 = N;  VGPR0 holds K = (lane>>4)*2, VGPR1 = K+1
//   C/D (16x16):    lane&15 = N;  VGPR r holds row M = r + (lane>>4)*8
// ---------------------------------------------------------------------------
__global__ __launch_bounds__(32 * WAVES_PER_BLOCK)
void gemm_pred_kernel(const float* __restrict__ x,
                      const float* __restrict__ W,
                      const float* __restrict__ Wb,
                      float* __restrict__ pred) {
  const int wave = threadIdx.x >> 5;
  const int lane = threadIdx.x & 31;

  const int tile  = blockIdx.x * WAVES_PER_BLOCK + wave; // 0 .. 20479
  const int igrp  = tile % IGRPS;
  const int otile = (tile / IGRPS) % OTILES;
  const int b     = tile / (IGRPS * OTILES);

  const int half = lane >> 4;        // 0 or 1
  const int ml   = lane & 15;        // M (rows of A) / N (cols of B)
  const int kb   = half << 1;        // K sub-offset: 0 or 2

  const float* __restrict__ Wrow = W + (size_t)(otile * 16 + ml) * CIN;
  const float* __restrict__ xcol =
      x + (size_t)b * CIN * NI + (igrp * 64 + ml);

  v8f acc0 = {}, acc1 = {}, acc2 = {}, acc3 = {};
#pragma unroll 2
  for (int k = 0; k < CIN; k += 4) {
    v2f a;
    a.x = Wrow[k + kb];
    a.y = Wrow[k + kb + 1];
    const float* __restrict__ p0 = xcol + (size_t)(k + kb) * NI;
    const float* __restrict__ p1 = p0 + NI;
    v2f b0, b1, b2, b3;
    b0.x = p0[0];  b0.y = p1[0];
    b1.x = p0[16]; b1.y = p1[16];
    b2.x = p0[32]; b2.y = p1[32];
    b3.x = p0[48]; b3.y = p1[48];
    acc0 = __builtin_amdgcn_wmma_f32_16x16x4_f32(false, a, false, b0,
                                                 (short)0, acc0, false, false);
    acc1 = __builtin_amdgcn_wmma_f32_16x16x4_f32(false, a, false, b1,
                                                 (short)0, acc1, false, false);
    acc2 = __builtin_amdgcn_wmma_f32_16x16x4_f32(false, a, false, b2,
                                                 (short)0, acc2, false, false);
    acc3 = __builtin_amdgcn_wmma_f32_16x16x4_f32(false, a, false, b3,
                                                 (short)0, acc3, false, false);
  }

  // Store D + bias: lane holds column i = igrp*64 + t*16 + (lane&15),
  // VGPR r is row o = otile*16 + (lane>>4)*8 + r.
  const int i0 = igrp * 64 + ml;
  const int rowbase = otile * 16 + half * 8;
  float* __restrict__ po = pred + ((size_t)b * NO) * NI + i0;
#pragma unroll
  for (int r = 0; r < 8; ++r) {
    const int o = rowbase + r;
    const float bias = Wb[o];
    float* __restrict__ prow = po + (size_t)o * NI;
    prow[0]  = acc0[r] + bias;
    prow[16] = acc1[r] + bias;
    prow[32] = acc2[r] + bias;
    prow[48] = acc3[r] + bias;
  }
}

// ---------------------------------------------------------------------------
// Kernel 2: dynamic routing, one workgroup (1024 threads) per batch element.
// Thread t owns in-capsule i = t; its routing logits b[j] stay in registers.
// s/v/norm live in LDS; reductions: wave32 shfl_xor tree + ds_add_f32 atomics.
// ---------------------------------------------------------------------------
__global__ __launch_bounds__(1024)
void routing_kernel(const float* __restrict__ pred, float* __restrict__ out) {
  const int b    = blockIdx.x;
  const int i    = threadIdx.x;       // 0..1023
  const int lane = threadIdx.x & 31;

  __shared__ float s_sh[NO];
  __shared__ float v_sh[NO];
  __shared__ float n2_sh[NJ];

  const float* __restrict__ predb = pred + (size_t)b * NO * NI;

  float blog[NJ];
#pragma unroll
  for (int j = 0; j < NJ; ++j) blog[j] = 0.0f;

  for (int it = 0; it < 3; ++it) {
    // c = softmax_j(blog)  (10 values, registers)
    float mx = blog[0];
#pragma unroll
    for (int j = 1; j < NJ; ++j) mx = fmaxf(mx, blog[j]);
    float c[NJ];
    float csum = 0.0f;
#pragma unroll
    for (int j = 0; j < NJ; ++j) { c[j] = __expf(blog[j] - mx); csum += c[j]; }
    const float cinv = 1.0f / csum;
#pragma unroll
    for (int j = 0; j < NJ; ++j) c[j] *= cinv;

    if (threadIdx.x < NO) s_sh[threadIdx.x] = 0.0f;
    if (threadIdx.x < NJ) n2_sh[threadIdx.x] = 0.0f;
    __syncthreads();

    // s[j][d] = sum_i c[j] * pred[j*D+d, i]   (wave tree + LDS atomic)
    for (int j = 0; j < NJ; ++j) {
      const float cj = c[j];
      const float* __restrict__ pj = predb + (size_t)(j * ND) * NI + i;
#pragma unroll
      for (int d = 0; d < ND; ++d) {
        float val = cj * pj[(size_t)d * NI];
#pragma unroll
        for (int off = 16; off > 0; off >>= 1)
          val += __shfl_xor(val, off, 32);
        if (lane == 0) atomicAdd(&s_sh[j * ND + d], val);
      }
    }
    __syncthreads();

    // squash: v = s * (|s|^2 / (1+|s|^2) / |s|), norm over d per capsule j
    if (threadIdx.x < NO) {
      const float sv = s_sh[threadIdx.x];
      atomicAdd(&n2_sh[threadIdx.x >> 5], sv * sv);   // ND == 32
    }
    __syncthreads();
    if (threadIdx.x < NO) {
      const float n2  = n2_sh[threadIdx.x >> 5];
      const float nrm = sqrtf(n2);
      v_sh[threadIdx.x] = s_sh[threadIdx.x] * (n2 / (1.0f + n2) / nrm);
    }
    __syncthreads();

    // blog[j] += sum_d v[j][d] * pred[j*D+d, i]
    for (int j = 0; j < NJ; ++j) {
      const float* __restrict__ pj = predb + (size_t)(j * ND) * NI + i;
      float db = 0.0f;
#pragma unroll
      for (int d = 0; d < ND; ++d)
        db += v_sh[j * ND + d] * pj[(size_t)d * NI];
      blog[j] += db;
    }
    __syncthreads();
  }

  // final v -> out[b, j, d]
  if (threadIdx.x < NO)
    out[(size_t)b * NO + threadIdx.x] = v_sh[threadIdx.x];
}

extern "C" void kernel_launch(void* const* d_in, const int* in_sizes, int n_in,
                              void* d_out, int out_size, void* d_ws, size_t ws_size,
                              hipStream_t stream) {
  const float* x  = (const float*)d_in[0];   // [64, 256, 32, 32]
  const float* W  = (const float*)d_in[1];   // [320, 256]
  const float* Wb = (const float*)d_in[2];   // [320]
  float* out  = (float*)d_out;               // [64, 10, 32]
  float* pred = (float*)d_ws;                // [64, 320, 1024] = 80 MB scratch

  const int tiles  = BSZ * OTILES * IGRPS;           // 20480 wave-tiles
  const int blocks = tiles / WAVES_PER_BLOCK;        // 2560

  gemm_pred_kernel<<<blocks, 32 * WAVES_PER_BLOCK, 0, stream>>>(x, W, Wb, pred);
  routing_kernel<<<BSZ, 1024, 0, stream>>>(pred, out);
}